// VectorQuantizer2_64707977281812
// MI455X (gfx1250) — compile-verified
//
#include <hip/hip_runtime.h>
#include <hip/hip_bf16.h>

typedef __attribute__((ext_vector_type(2))) float v2f;
typedef __attribute__((ext_vector_type(8))) float v8f;

#define CD   128          // channels
#define BD   32           // batch
#define HD   16           // H == W
#define VD   8192         // vocab
#define NUMEL (BD*CD*HD*HD)   // 1048576
#define NPART 2560            // 5 scales * 512 conv blocks

__device__ __forceinline__ v8f wmma4(v2f a, v2f b, v8f c) {
  // D(16x16,f32) = A(16x4,f32) * B(4x16,f32) + C
  return __builtin_amdgcn_wmma_f32_16x16x4_f32(false, a, false, b, (short)0, c, false, false);
}

// ---------------- prep kernels ----------------
__global__ void k_esq(const float* __restrict__ emb, float* __restrict__ esq) {
  int v = blockIdx.x * blockDim.x + threadIdx.x;
  if (v >= VD) return;
  const float* e = emb + (size_t)v * CD;
  float s = 0.f;
  for (int c = 0; c < CD; ++c) s += e[c] * e[c];
  esq[v] = s;
}

// wt2[kphi][kk][n] = (W[2kk][n], W[2kk+1][n]),  W[k][n] = phi_w[kphi][n][k%128][ (k/128)/3 ][ (k/128)%3 ]
__global__ void k_wt2(const float* __restrict__ phi_w, v2f* __restrict__ wt2) {
  int tid = blockIdx.x * blockDim.x + threadIdx.x;
  const int total = 4 * 576 * CD;
  if (tid >= total) return;
  int kphi = tid / (576 * CD);
  int rem  = tid % (576 * CD);
  int kk   = rem / CD;
  int n    = rem % CD;
  int k    = 2 * kk;
  int cin  = k % CD;        // even
  int tap  = k / CD;        // 0..8
  const float* base = phi_w + (((size_t)kphi * CD + n) * CD) * 9;
  v2f w;
  w.x = base[(size_t)cin * 9 + tap];
  w.y = base[(size_t)(cin + 1) * 9 + tap];
  wt2[tid] = w;
}

__global__ void k_init(const float* __restrict__ f, float* __restrict__ f_rest,
                       float* __restrict__ f_hat, float* __restrict__ partials) {
  for (int i = blockIdx.x * blockDim.x + threadIdx.x; i < NUMEL; i += gridDim.x * blockDim.x) {
    f_rest[i] = f[i];
    f_hat[i]  = 0.f;
    if (i < NPART) partials[i] = 0.f;
  }
}

// ---------------- per-scale kernels ----------------
// rest_NC[n][c], n = b*pn*pn + py*pn + px  (area pooling of f_rest)
__global__ void k_pool(const float* __restrict__ f_rest, float* __restrict__ xnc, int pn) {
  int total = BD * pn * pn * CD;
  int tid = blockIdx.x * blockDim.x + threadIdx.x;
  if (tid >= total) return;
  int n = tid / CD, c = tid % CD;
  int b = n / (pn * pn);
  int p = n % (pn * pn);
  int py = p / pn, px = p % pn;
  int s = HD / pn;
  const float* base = f_rest + ((size_t)(b * CD + c) * HD + py * s) * HD + px * s;
  float acc = 0.f;
  for (int i = 0; i < s; ++i)
    for (int j = 0; j < s; ++j)
      acc += base[i * HD + j];
  xnc[tid] = acc * (1.f / (float)(s * s));
}

// nearest-codebook search via f32 WMMA.
// One block = 16 rows; 4 waves split V; each wave processes TWO 16-column code
// tiles per iteration with independent accumulator chains (ILP for the XDL pipe,
// A-fragments in registers reused across both tiles).
__global__ void k_argmin(const float* __restrict__ xnc, const float* __restrict__ emb,
                         const float* __restrict__ esq, int* __restrict__ idx) {
  __shared__ float cd[4][16];
  __shared__ int   cv[4][16];
  const int wave = threadIdx.x >> 5;
  const int lane = threadIdx.x & 31;
  const int m    = lane & 15;
  const int half = lane >> 4;
  const int row  = blockIdx.x * 16 + m;

  // A fragments for the whole K=128 (32 k-steps), kept in registers
  v2f a[32];
  const float* xr = xnc + (size_t)row * CD + 2 * half;
#pragma unroll
  for (int j = 0; j < 32; ++j) a[j] = *(const v2f*)(xr + j * 4);

  float dmin[8];
  int   vmin[8];
#pragma unroll
  for (int r = 0; r < 8; ++r) { dmin[r] = 3.4e38f; vmin[r] = 0; }

  const int vper = VD / 4;   // 2048 codes per wave
  const int vbeg = wave * vper;
  for (int v0 = vbeg; v0 < vbeg + vper; v0 += 32) {
    v8f acc0 = {};
    v8f acc1 = {};
    const float* ep0 = emb + (size_t)(v0 + m) * CD + 2 * half;
    const float* ep1 = ep0 + 16 * CD;
#pragma unroll
    for (int j = 0; j < 32; ++j) {
      v2f b0 = *(const v2f*)(ep0 + j * 4);
      v2f b1 = *(const v2f*)(ep1 + j * 4);
      acc0 = wmma4(a[j], b0, acc0);
      acc1 = wmma4(a[j], b1, acc1);
    }
    float es0 = esq[v0 + m];
    float es1 = esq[v0 + 16 + m];
#pragma unroll
    for (int r = 0; r < 8; ++r) {
      float d0 = es0 - 2.f * acc0[r];
      if (d0 < dmin[r]) { dmin[r] = d0; vmin[r] = v0 + m; }
      float d1 = es1 - 2.f * acc1[r];
      if (d1 < dmin[r]) { dmin[r] = d1; vmin[r] = v0 + 16 + m; }
    }
  }
  // reduce across the 16 lanes of each half (xor masks < 16 stay in-half)
  for (int off = 1; off < 16; off <<= 1) {
#pragma unroll
    for (int r = 0; r < 8; ++r) {
      float d2 = __shfl_xor(dmin[r], off, 32);
      int   v2_ = __shfl_xor(vmin[r], off, 32);
      if (d2 < dmin[r] || (d2 == dmin[r] && v2_ < vmin[r])) { dmin[r] = d2; vmin[r] = v2_; }
    }
  }
  if (m == 0) {
#pragma unroll
    for (int r = 0; r < 8; ++r) { cd[wave][half * 8 + r] = dmin[r]; cv[wave][half * 8 + r] = vmin[r]; }
  }
  __syncthreads();
  if (threadIdx.x < 16) {
    int t = threadIdx.x;
    float bd = cd[0][t]; int bv = cv[0][t];
    for (int w = 1; w < 4; ++w) {
      float d2 = cd[w][t]; int v2_ = cv[w][t];
      if (d2 < bd || (d2 == bd && v2_ < bv)) { bd = d2; bv = v2_; }
    }
    idx[blockIdx.x * 16 + t] = bv;
  }
}

__device__ __forceinline__ float keys_cubic(float t) {
  float at = fabsf(t);
  if (at <= 1.f)      return (1.5f * at - 2.5f) * at * at + 1.f;
  else if (at < 2.f)  return ((-0.5f * at + 2.5f) * at - 4.f) * at + 2.f;
  return 0.f;
}

// h_up[b][c][y][x] = bicubic upsample of embedding[idx] (pn x pn -> 16 x 16)
__global__ void k_upsample(const float* __restrict__ emb, const int* __restrict__ idx,
                           float* __restrict__ h_up, int pn) {
  for (int i = blockIdx.x * blockDim.x + threadIdx.x; i < NUMEL; i += gridDim.x * blockDim.x) {
    int b = i >> 15;
    int c = (i >> 8) & 127;
    int y = (i >> 4) & 15;
    int x = i & 15;
    float out;
    if (pn == HD) {
      out = emb[(size_t)idx[(b * HD + y) * HD + x] * CD + c];
    } else {
      float scale = (float)pn * (1.f / (float)HD);
      float sfy = ((float)y + 0.5f) * scale - 0.5f;
      float sfx = ((float)x + 0.5f) * scale - 0.5f;
      int fy = (int)floorf(sfy), fx = (int)floorf(sfx);
      float wy[4], wx[4]; int iy[4], ix[4];
      int ny = 0, nx = 0;
      float wsy = 0.f, wsx = 0.f;
      for (int t = -1; t <= 2; ++t) {
        int yy = fy + t;
        if (yy >= 0 && yy < pn) {
          float w = keys_cubic(sfy - (float)yy);
          wy[ny] = w; iy[ny] = yy; ++ny; wsy += w;
        }
        int xx = fx + t;
        if (xx >= 0 && xx < pn) {
          float w = keys_cubic(sfx - (float)xx);
          wx[nx] = w; ix[nx] = xx; ++nx; wsx += w;
        }
      }
      float acc = 0.f;
      for (int a = 0; a < ny; ++a) {
        float rowacc = 0.f;
        for (int e = 0; e < nx; ++e)
          rowacc += wx[e] * emb[(size_t)idx[(b * pn + iy[a]) * pn + ix[e]] * CD + c];
        acc += wy[a] * rowacc;
      }
      out = acc / (wsy * wsx);
    }
    h_up[i] = out;
  }
}

// fused: y = conv3x3(h_up)+bias ; h = 0.5*h_up + 0.5*y ; f_hat += h ; f_rest -= h ; partial += (f_hat-f)^2
// one block per (b, y) output row: 16 pixels x 128 couts via 8 waves of WMMA over K = 9*128,
// two independent accumulator chains (even/odd k-steps) for ILP.
__global__ void k_conv_phi(const float* __restrict__ h_up, const v2f* __restrict__ wt2,
                           const float* __restrict__ phi_b, const float* __restrict__ f,
                           float* __restrict__ f_hat, float* __restrict__ f_rest,
                           float* __restrict__ partials, int kphi, int si) {
  __shared__ float s_in[3 * 18 * CD];   // [r][col][c] halo tile, 27648 B
  __shared__ float s_red[256];
  const int bid = blockIdx.x;
  const int b = bid >> 4, y = bid & 15;
  const float* base = h_up + (size_t)b * CD * HD * HD;
  for (int t = threadIdx.x; t < 3 * 18 * CD; t += 256) {
    int rc = t >> 7, c = t & 127;
    int r = rc / 18, col = rc % 18;
    int gy = y + r - 1, gx = col - 1;
    float v = 0.f;
    if (gy >= 0 && gy < HD && gx >= 0 && gx < HD) v = base[c * (HD * HD) + gy * HD + gx];
    s_in[t] = v;
  }
  __syncthreads();

  const int wave = threadIdx.x >> 5;
  const int lane = threadIdx.x & 31;
  const int m    = lane & 15;     // pixel x for A-rows, cout offset for B-cols
  const int half = lane >> 4;
  const int cout = wave * 16 + m;

  v8f acc0 = {};
  v8f acc1 = {};
  const v2f* wbase = wt2 + (size_t)kphi * 576 * CD;
  for (int tap = 0; tap < 9; ++tap) {
    int dy = tap / 3, dx = tap % 3;
    const float* srow = s_in + (dy * 18 + m + dx) * CD + 2 * half;
    const v2f* wrow = wbase + (size_t)(tap * 64 + half) * CD + m;
#pragma unroll
    for (int j = 0; j < 32; j += 2) {
      v2f a0 = *(const v2f*)(srow + j * 4);
      v2f a1 = *(const v2f*)(srow + j * 4 + 4);
      v2f b0 = wrow[(size_t)j * 256];
      v2f b1 = wrow[(size_t)(j + 1) * 256];
      acc0 = wmma4(a0, b0, acc0);
      acc1 = wmma4(a1, b1, acc1);
    }
  }

  float bias = phi_b[kphi * CD + cout];
  float local = 0.f;
#pragma unroll
  for (int r = 0; r < 8; ++r) {
    int x = r + half * 8;
    int i = (b * CD + cout) * (HD * HD) + y * HD + x;
    float hval = 0.5f * h_up[i] + 0.5f * (acc0[r] + acc1[r] + bias);
    float fh = f_hat[i] + hval;
    f_hat[i] = fh;
    f_rest[i] -= hval;
    float df = fh - f[i];
    local += df * df;
  }
  s_red[threadIdx.x] = local;
  __syncthreads();
  for (int s = 128; s > 0; s >>= 1) {
    if (threadIdx.x < s) s_red[threadIdx.x] += s_red[threadIdx.x + s];
    __syncthreads();
  }
  if (threadIdx.x == 0) partials[si * 512 + bid] = s_red[0];
}

// ---------------- finalize ----------------
__global__ void k_copy(const float* __restrict__ f_hat, float* __restrict__ out) {
  for (int i = blockIdx.x * blockDim.x + threadIdx.x; i < NUMEL; i += gridDim.x * blockDim.x)
    out[i] = f_hat[i];   // straight-through output == f_hat numerically
}

__global__ void k_loss(const float* __restrict__ partials, float* __restrict__ out) {
  __shared__ float s_red[256];
  float s = 0.f;
  for (int i = threadIdx.x; i < NPART; i += 256) s += partials[i];
  s_red[threadIdx.x] = s;
  __syncthreads();
  for (int st = 128; st > 0; st >>= 1) {
    if (threadIdx.x < st) s_red[threadIdx.x] += s_red[threadIdx.x + st];
    __syncthreads();
  }
  // loss = (1/5) * sum_si (beta+1)*mean = 1.25/5/NUMEL * total
  if (threadIdx.x == 0) out[NUMEL] = s_red[0] * (1.25f / (5.f * (float)NUMEL));
}

extern "C" void kernel_launch(void* const* d_in, const int* in_sizes, int n_in,
                              void* d_out, int out_size, void* d_ws, size_t ws_size,
                              hipStream_t stream) {
  const float* f     = (const float*)d_in[0];   // [32,128,16,16]
  const float* emb   = (const float*)d_in[1];   // [8192,128]
  const float* phi_w = (const float*)d_in[2];   // [4,128,128,3,3]
  const float* phi_b = (const float*)d_in[3];   // [4,128]
  float* out = (float*)d_out;

  char* ws = (char*)d_ws;
  size_t off = 0;
  auto carve = [&](size_t bytes) -> void* {
    void* p = ws + off;
    off = (off + bytes + 255) & ~(size_t)255;
    return p;
  };
  float* f_rest   = (float*)carve((size_t)NUMEL * 4);
  float* f_hat    = (float*)carve((size_t)NUMEL * 4);
  float* xnc      = (float*)carve((size_t)VD * CD * 4);   // max N = 8192 rows
  float* h_up     = (float*)carve((size_t)NUMEL * 4);
  float* esq      = (float*)carve((size_t)VD * 4);
  v2f*   wt2      = (v2f*)  carve((size_t)4 * 576 * CD * 8);
  int*   idxbuf   = (int*)  carve((size_t)VD * 4);
  float* partials = (float*)carve((size_t)NPART * 4);
  (void)ws_size; (void)in_sizes; (void)n_in; (void)out_size;

  k_esq <<<VD / 256, 256, 0, stream>>>(emb, esq);
  k_wt2 <<<(4 * 576 * CD + 255) / 256, 256, 0, stream>>>(phi_w, wt2);
  k_init<<<1024, 256, 0, stream>>>(f, f_rest, f_hat, partials);

  const int pns[5]   = {1, 2, 4, 8, 16};
  const int kphis[5] = {0, 1, 1, 2, 3};   // nearest tick of linspace(1/12,11/12,4), first-on-tie
  for (int si = 0; si < 5; ++si) {
    int pn = pns[si];
    int N = BD * pn * pn;
    int total = N * CD;
    k_pool   <<<(total + 255) / 256, 256, 0, stream>>>(f_rest, xnc, pn);
    k_argmin <<<N / 16, 128, 0, stream>>>(xnc, emb, esq, idxbuf);
    k_upsample<<<1024, 256, 0, stream>>>(emb, idxbuf, h_up, pn);
    k_conv_phi<<<BD * HD, 256, 0, stream>>>(h_up, wt2, phi_b, f, f_hat, f_rest,
                                            partials, kphis[si], si);
  }
  k_copy<<<1024, 256, 0, stream>>>(f_hat, out);
  k_loss<<<1, 256, 0, stream>>>(partials, out);
}